// SwavBase_55370718380494
// MI455X (gfx1250) — compile-verified
//
#include <hip/hip_runtime.h>

// Problem sizes fixed by reference setup_inputs()
#define NROWS 131072
#define DDIM  64
#define KPROT 512
#define MTILE 32
#define GEMM_BLOCKS (NROWS / MTILE)   // 4096

typedef __attribute__((ext_vector_type(2))) float v2f;
typedef __attribute__((ext_vector_type(8))) float v8f;

#define FLT_BIG 3.402823466e+38f

// ---------------- init column-min accumulator to +inf bits ----------------
__global__ void k_init(unsigned int* __restrict__ gcol) {
  int t = blockIdx.x * blockDim.x + threadIdx.x;
  if (t < KPROT) gcol[t] = 0x7F800000u;
}

// ---------------- L2 normalize, one wave32 per row ----------------
__global__ void k_norm(const float* __restrict__ x, float* __restrict__ xn0,
                       float* __restrict__ xn1, float* __restrict__ a2) {
  int row  = (blockIdx.x * blockDim.x + threadIdx.x) >> 5;
  int lane = threadIdx.x & 31;
  if (row >= NROWS) return;
  float2 v = ((const float2*)(x + (size_t)row * DDIM))[lane];
  float s = v.x * v.x + v.y * v.y;
  s += __shfl_xor(s, 16, 32);
  s += __shfl_xor(s, 8, 32);
  s += __shfl_xor(s, 4, 32);
  s += __shfl_xor(s, 2, 32);
  s += __shfl_xor(s, 1, 32);
  float inv = 1.0f / fmaxf(sqrtf(s), 1e-12f);
  float2 o; o.x = v.x * inv; o.y = v.y * inv;
  ((float2*)(xn0 + (size_t)row * DDIM))[lane] = o;
  ((float2*)(xn1 + (size_t)row * DDIM))[lane] = o;
  if (lane == 0) a2[row] = s * inv * inv;
}

// ---------------- ||W_k||^2, one wave32 per prototype ----------------
__global__ void k_wnorm(const float* __restrict__ W, float* __restrict__ b2) {
  int row  = (blockIdx.x * blockDim.x + threadIdx.x) >> 5;
  int lane = threadIdx.x & 31;
  if (row >= KPROT) return;
  float2 v = ((const float2*)(W + (size_t)row * DDIM))[lane];
  float s = v.x * v.x + v.y * v.y;
  s += __shfl_xor(s, 16, 32);
  s += __shfl_xor(s, 8, 32);
  s += __shfl_xor(s, 4, 32);
  s += __shfl_xor(s, 2, 32);
  s += __shfl_xor(s, 1, 32);
  if (lane == 0) b2[row] = s;
}

// ---------------- fused GEMM (WMMA f32 16x16x4) + cdist + min reductions ----
// Block: 256 threads = 8 waves. Block covers 32 rows x all 512 columns.
// Waves 0-3: rows [0,16), col groups 0..3 (8 column tiles of 16 each).
// Waves 4-7: rows [16,32), col groups 0..3.
__global__ void __launch_bounds__(256) k_gemm(
    const float* __restrict__ xn, const float* __restrict__ W,
    const float* __restrict__ a2, const float* __restrict__ b2,
    float* __restrict__ proto, float* __restrict__ rowpart,
    unsigned int* __restrict__ gcol) {
  __shared__ float        rmin_lds[2][16][4];
  __shared__ float        rsum_lds[32];
  __shared__ unsigned int cmin_lds[KPROT];

  const int tid    = threadIdx.x;
  const int wave   = tid >> 5;
  const int lane   = tid & 31;
  const int rowgrp = wave >> 2;          // 0..1
  const int colgrp = wave & 3;           // 0..3
  const int l15    = lane & 15;
  const int hi     = lane >> 4;          // 0 or 1
  const int off    = hi * 2;             // depth sub-offset for A/B fragments
  const int m0     = blockIdx.x * MTILE + rowgrp * 16;
  const int nbase  = colgrp * 128 + l15; // this wave's 8 tiles: n = nbase + tt*16

  // init block-local column-min (+inf bits)
  cmin_lds[tid]       = 0x7F800000u;
  cmin_lds[tid + 256] = 0x7F800000u;
  __syncthreads();

  v8f acc[8];
#pragma unroll
  for (int tt = 0; tt < 8; ++tt) {
#pragma unroll
    for (int r = 0; r < 8; ++r) acc[tt][r] = 0.0f;
  }

  // A fragment base: lane l15 -> M row, depth index = kd + off + {0,1}
  const float* arow = xn + (size_t)(m0 + l15) * DDIM + off;

#pragma unroll
  for (int kd = 0; kd < DDIM; kd += 4) {
    float2 av = *(const float2*)(arow + kd);
    v2f a; a.x = av.x; a.y = av.y;
#pragma unroll
    for (int tt = 0; tt < 8; ++tt) {
      int n = nbase + tt * 16;
      // B[depth][n] = W[n][depth]; lane l15 -> column n, regs -> depth kd+off+{0,1}
      float2 bv = *(const float2*)(W + (size_t)n * DDIM + kd + off);
      v2f b; b.x = bv.x; b.y = bv.y;
      acc[tt] = __builtin_amdgcn_wmma_f32_16x16x4_f32(
          false, a, false, b, (short)0, acc[tt], false, false);
    }
  }

  // ---- epilogue: proto store + fused cdist + min reductions ----
  float a2v[8];
#pragma unroll
  for (int r = 0; r < 8; ++r) a2v[r] = a2[m0 + hi * 8 + r];

  float rm[8];
#pragma unroll
  for (int r = 0; r < 8; ++r) rm[r] = FLT_BIG;

#pragma unroll
  for (int tt = 0; tt < 8; ++tt) {
    int n = nbase + tt * 16;
    float b2v = b2[n];
    float cmin = FLT_BIG;
#pragma unroll
    for (int r = 0; r < 8; ++r) {
      float dot = acc[tt][r];
      proto[(size_t)(m0 + hi * 8 + r) * KPROT + n] = dot;
      float sq = a2v[r] + b2v - 2.0f * dot;
      float d  = sqrtf(fmaxf(sq, 1e-12f));
      rm[r]    = fminf(rm[r], d);
      cmin     = fminf(cmin, d);
    }
    // lanes l and l^16 hold the same column n for different row halves
    cmin = fminf(cmin, __shfl_xor(cmin, 16, 32));
    if (hi == 0) atomicMin(&cmin_lds[n], __float_as_uint(cmin));
  }

  // row mins across the 16 lanes of each half-wave
#pragma unroll
  for (int r = 0; r < 8; ++r) {
    float v = rm[r];
    v = fminf(v, __shfl_xor(v, 1, 32));
    v = fminf(v, __shfl_xor(v, 2, 32));
    v = fminf(v, __shfl_xor(v, 4, 32));
    v = fminf(v, __shfl_xor(v, 8, 32));
    if (l15 == 0) rmin_lds[rowgrp][hi * 8 + r][colgrp] = v;  // lanes 0 and 16
  }
  __syncthreads();

  // block -> global column mins (exact, order-independent => deterministic)
  atomicMin(&gcol[tid],       cmin_lds[tid]);
  atomicMin(&gcol[tid + 256], cmin_lds[tid + 256]);

  // combine row mins across the 4 column groups, then sum 32 rows
  if (tid < 32) {
    int rg = tid >> 4, ml = tid & 15;
    float v = fminf(fminf(rmin_lds[rg][ml][0], rmin_lds[rg][ml][1]),
                    fminf(rmin_lds[rg][ml][2], rmin_lds[rg][ml][3]));
    rsum_lds[tid] = v;
  }
  __syncthreads();
  if (tid == 0) {
    float s = 0.0f;
    for (int i = 0; i < 32; ++i) s += rsum_lds[i];
    rowpart[blockIdx.x] = s;   // fixed-order partial => deterministic
  }
}

// ---------------- final deterministic reduction + scalar losses ----------------
__global__ void k_final(const float* __restrict__ rowpart,
                        const unsigned int* __restrict__ gcol,
                        const float* __restrict__ recon,
                        const float* __restrict__ kl,
                        const float* __restrict__ mmd,
                        float* __restrict__ out_scalars) {
  __shared__ float sdata[512];
  int t = threadIdx.x;

  float s = 0.0f;
  for (int i = t; i < GEMM_BLOCKS; i += 512) s += rowpart[i];
  sdata[t] = s;
  __syncthreads();
  for (int w = 256; w >= 1; w >>= 1) {
    if (t < w) sdata[t] += sdata[t + w];
    __syncthreads();
  }
  float rowsum = sdata[0];
  __syncthreads();

  sdata[t] = __uint_as_float(gcol[t]);
  __syncthreads();
  for (int w = 256; w >= 1; w >>= 1) {
    if (t < w) sdata[t] += sdata[t + w];
    __syncthreads();
  }
  if (t == 0) {
    float colsum = sdata[0];
    float prot = 0.5f * (rowsum / (float)NROWS) + 0.5f * (colsum / (float)KPROT);
    float cvae = recon[0] + 0.5f * kl[0] + mmd[0];
    out_scalars[0] = cvae;
    out_scalars[1] = prot;
  }
}

extern "C" void kernel_launch(void* const* d_in, const int* in_sizes, int n_in,
                              void* d_out, int out_size, void* d_ws, size_t ws_size,
                              hipStream_t stream) {
  const float* x     = (const float*)d_in[0];
  const float* W     = (const float*)d_in[1];
  const float* recon = (const float*)d_in[2];
  const float* kl    = (const float*)d_in[3];
  const float* mmd   = (const float*)d_in[4];

  float* out   = (float*)d_out;
  float* xn0   = out;                                      // [N*D]
  float* xn1   = out + (size_t)NROWS * DDIM;               // [N*D]
  float* proto = out + 2 * (size_t)NROWS * DDIM;           // [N*K]
  float* outsc = proto + (size_t)NROWS * KPROT;            // [2] cvae, prot

  // workspace layout (floats)
  float* ws_f           = (float*)d_ws;
  float* a2             = ws_f;                            // [N]
  float* b2             = ws_f + NROWS;                    // [K]
  unsigned int* gcol    = (unsigned int*)(ws_f + NROWS + KPROT);  // [K]
  float* rowpart        = ws_f + NROWS + 2 * KPROT;        // [GEMM_BLOCKS]

  k_init<<<2, 256, 0, stream>>>(gcol);
  k_norm<<<NROWS / 8, 256, 0, stream>>>(x, xn0, xn1, a2);
  k_wnorm<<<KPROT / 8, 256, 0, stream>>>(W, b2);
  k_gemm<<<GEMM_BLOCKS, 256, 0, stream>>>(xn0, W, a2, b2, proto, rowpart, gcol);
  k_final<<<1, 512, 0, stream>>>(rowpart, gcol, recon, kl, mmd, outsc);
}